// ModifiedGATv2_20469814132986
// MI455X (gfx1250) — compile-verified
//
#include <hip/hip_runtime.h>

// Problem constants (from reference)
#define NNODES   512          // VOCAB
#define INFEAT   512
#define CONDF    512
#define NHEADS   8
#define FDIM     32           // N_HIDDEN / N_HEADS
#define RDIM     256          // N_HEADS * FDIM
#define PDIM     1024         // 2 * IN_FEAT
#define BATCH    2
#define SLOPE    0.2f
#define NEG_INF_V (-1000000000.0f)

typedef __attribute__((ext_vector_type(2))) float v2f;
typedef __attribute__((ext_vector_type(8))) float v8f;

// ---------------------------------------------------------------------------
// Generic fp32 WMMA GEMM:  D[M x N] = A[M x K] @ Bt[N x K]^T
// A rows optionally gathered through `gather` (emb_table[x[i]]).
// One wave per 16x16 output tile; V_WMMA_F32_16X16X4_F32, K-step = 4.
// Both A and Bt are contiguous along K -> 8-byte aligned b64 fragment loads.
// ---------------------------------------------------------------------------
__global__ __launch_bounds__(256) void wmma_gemm_abt(
    const float* __restrict__ A, int lda,
    const float* __restrict__ Bt, int ldb,
    float* __restrict__ D, int ldd,
    int tilesM, int tilesN, int K,
    const int* __restrict__ gather)
{
    int wave = threadIdx.x >> 5;
    int tile = blockIdx.x * 8 + wave;
    if (tile >= tilesM * tilesN) return;           // wave-uniform: EXEC stays full
    int tm = tile / tilesN;
    int tn = tile % tilesN;

    int lane = threadIdx.x & 31;
    int m    = lane & 15;
    int koff = (lane < 16) ? 0 : 2;                // A/B frag: lanes 0-15 -> K 0,1 ; 16-31 -> K 2,3

    int arow = tm * 16 + m;
    if (gather) arow = gather[arow];
    const float* Ap = A  + (size_t)arow * lda + koff;
    const float* Bp = Bt + (size_t)(tn * 16 + m) * ldb + koff;

    v8f acc = {};
    for (int k = 0; k < K; k += 4) {
        v2f av = *(const v2f*)(Ap + k);
        v2f bv = *(const v2f*)(Bp + k);
        acc = __builtin_amdgcn_wmma_f32_16x16x4_f32(
            false, av, false, bv, (short)0, acc, false, false);
    }

    // C/D layout: VGPR v -> row v (lanes 0-15) / row v+8 (lanes 16-31), col = lane&15
    int rowHalf = (lane < 16) ? 0 : 8;
    float* Dp = D + (size_t)(tm * 16 + rowHalf) * ldd + tn * 16 + m;
#pragma unroll
    for (int v = 0; v < 8; ++v)
        Dp[(size_t)v * ldd] = acc[v];
}

// ---------------------------------------------------------------------------
// t1[b][p] = proj_b[p] + sum_c proj_w[p][512+c] * cond[b][c]       (2 x 1024)
// ---------------------------------------------------------------------------
__global__ void cond_proj(const float* __restrict__ proj_w,
                          const float* __restrict__ proj_b,
                          const float* __restrict__ cond,
                          float* __restrict__ t1)
{
    int idx = blockIdx.x * blockDim.x + threadIdx.x;
    if (idx >= BATCH * PDIM) return;
    int b = idx / PDIM, p = idx % PDIM;
    const float* wr = proj_w + (size_t)p * PDIM + INFEAT;
    const float* c  = cond + (size_t)b * CONDF;
    float s = proj_b[p];
    for (int k = 0; k < CONDF; ++k) s = fmaf(wr[k], c[k], s);
    t1[idx] = s;
}

// ---------------------------------------------------------------------------
// cvec[b][r] = sum_p lin_w[r][p] * t1[b][p]                         (2 x 256)
// ---------------------------------------------------------------------------
__global__ void cond_lin(const float* __restrict__ lin_w,
                         const float* __restrict__ t1,
                         float* __restrict__ cvec)
{
    int idx = blockIdx.x * blockDim.x + threadIdx.x;
    if (idx >= BATCH * RDIM) return;
    int b = idx / RDIM, r = idx % RDIM;
    const float* wr = lin_w + (size_t)r * PDIM;
    const float* t  = t1 + (size_t)b * PDIM;
    float s = 0.f;
    for (int p = 0; p < PDIM; ++p) s = fmaf(wr[p], t[p], s);
    cvec[idx] = s;
}

// ---------------------------------------------------------------------------
// g[b][i][r] = gsh[i][r] + cvec[b][r]
// ---------------------------------------------------------------------------
__global__ void add_cond(const float* __restrict__ gsh,
                         const float* __restrict__ cvec,
                         float* __restrict__ g)
{
    int idx = blockIdx.x * blockDim.x + threadIdx.x;
    if (idx >= BATCH * NNODES * RDIM) return;
    int b   = idx / (NNODES * RDIM);
    int rem = idx - b * (NNODES * RDIM);
    int r   = rem % RDIM;
    g[idx] = gsh[rem] + cvec[b * RDIM + r];
}

// ---------------------------------------------------------------------------
// Fused GATv2 attention: one workgroup per (b, h, 16-row i-tile).
//  - stage g[b,:,h,:] (512x32, LDS stride 33) once
//  - wave w computes score row i = it*16+w : leaky-relu dot, mask, softmax
//    (register-resident 16 scores/lane + __shfl_xor wave reductions)
//  - waves 0,1 run the (16x512)@(512x32) aggregation with f32 WMMA from LDS
// Dynamic LDS: 512*33 + 16*513 + 32 floats = ~100.5 KB (<< 320 KB/WGP)
// ---------------------------------------------------------------------------
#define GH_STRIDE 33
#define S_STRIDE  513

__global__ __launch_bounds__(512) void gat_attn(
    const float* __restrict__ g,
    const int*   __restrict__ adj,
    const float* __restrict__ attn_w,
    float* __restrict__ out)
{
    extern __shared__ float smem[];
    float* Gh  = smem;                                 // [512][33]
    float* S   = smem + NNODES * GH_STRIDE;            // [16][513]
    float* Wsh = S + 16 * S_STRIDE;                    // [32]

    int it = blockIdx.x, h = blockIdx.y, b = blockIdx.z;
    int tid = threadIdx.x;

    const float* gb = g + (size_t)b * NNODES * RDIM + h * FDIM;

    // Stage the head slice of g into LDS (padded rows to avoid bank conflicts)
    for (int idx = tid; idx < NNODES * FDIM; idx += 512) {
        int j = idx >> 5, f = idx & 31;
        Gh[j * GH_STRIDE + f] = gb[(size_t)j * RDIM + f];
    }
    if (tid < FDIM) Wsh[tid] = attn_w[tid];
    __syncthreads();

    int w = tid >> 5;          // wave id = local row
    int l = tid & 31;          // lane
    int i = it * 16 + w;       // global row

    // ---- scores: lane l owns j = jj*32 + l, jj = 0..15 ----
    float ereg[16];
    const float* Gi = Gh + i * GH_STRIDE;
#pragma unroll
    for (int jj = 0; jj < 16; ++jj) {
        int j = jj * 32 + l;
        const float* Gj = Gh + j * GH_STRIDE;
        float acc = 0.f;
#pragma unroll
        for (int f = 0; f < FDIM; ++f) {
            float s = Gi[f] + Gj[f];
            s = (s >= 0.f) ? s : SLOPE * s;
            acc = fmaf(s, Wsh[f], acc);
        }
        ereg[jj] = adj[(size_t)i * NNODES + j] ? acc : NEG_INF_V;
    }

    // ---- masked softmax over the 512 j's of row i ----
    float mx = ereg[0];
#pragma unroll
    for (int jj = 1; jj < 16; ++jj) mx = fmaxf(mx, ereg[jj]);
    for (int off = 16; off >= 1; off >>= 1)
        mx = fmaxf(mx, __shfl_xor(mx, off, 32));

    float preg[16];
    float se = 0.f;
#pragma unroll
    for (int jj = 0; jj < 16; ++jj) {
        preg[jj] = __expf(ereg[jj] - mx);
        se += preg[jj];
    }
    for (int off = 16; off >= 1; off >>= 1)
        se += __shfl_xor(se, off, 32);
    float inv = 1.0f / se;

#pragma unroll
    for (int jj = 0; jj < 16; ++jj)
        S[w * S_STRIDE + jj * 32 + l] = preg[jj] * inv;
    __syncthreads();

    // ---- aggregation: out_tile(16x32) = S(16x512) @ Gh(512x32) via f32 WMMA ----
    if (w < 2) {                                   // wave-uniform branch, EXEC full
        int lo   = (l < 16);
        int mrow = l & 15;
        int koff = lo ? 0 : 2;
        int n    = w * 16 + mrow;                  // output column within head dim

        v8f acc = {};
        for (int k = 0; k < NNODES; k += 4) {
            v2f av, bv;
            av.x = S[mrow * S_STRIDE + k + koff];
            av.y = S[mrow * S_STRIDE + k + koff + 1];
            bv.x = Gh[(k + koff)     * GH_STRIDE + n];
            bv.y = Gh[(k + koff + 1) * GH_STRIDE + n];
            acc = __builtin_amdgcn_wmma_f32_16x16x4_f32(
                false, av, false, bv, (short)0, acc, false, false);
        }
        int rowHalf = lo ? 0 : 8;
        float* ob = out + ((size_t)b * NNODES + it * 16 + rowHalf) * RDIM + h * FDIM + n;
#pragma unroll
        for (int v = 0; v < 8; ++v)
            ob[(size_t)v * RDIM] = acc[v];
    }
}

// ---------------------------------------------------------------------------
// Launch
// inputs: 0:x(int 512) 1:adj(int 512*512) 2:cond(f32 2*512) 3:emb(f32 512*512)
//         4:proj_w(f32 1024*1024) 5:proj_b(f32 1024) 6:lin_w(f32 256*1024)
//         7:attn_w(f32 32)      output: f32 2*512*256
// ---------------------------------------------------------------------------
extern "C" void kernel_launch(void* const* d_in, const int* in_sizes, int n_in,
                              void* d_out, int out_size, void* d_ws, size_t ws_size,
                              hipStream_t stream) {
    const int*   x      = (const int*)  d_in[0];
    const int*   adj    = (const int*)  d_in[1];
    const float* cond   = (const float*)d_in[2];
    const float* emb    = (const float*)d_in[3];
    const float* proj_w = (const float*)d_in[4];
    const float* proj_b = (const float*)d_in[5];
    const float* lin_w  = (const float*)d_in[6];
    const float* attn_w = (const float*)d_in[7];
    float* out = (float*)d_out;

    float* ws   = (float*)d_ws;
    float* Hn   = ws;                          // 512*1024
    float* gsh  = Hn  + NNODES * PDIM;         // 512*256
    float* t1   = gsh + NNODES * RDIM;         // 2*1024
    float* cvec = t1  + BATCH * PDIM;          // 2*256
    float* gg   = cvec + BATCH * RDIM;         // 2*512*256

    // K1: Hn = emb[x] @ proj_w[:, :512]^T   (M=512, N=1024, K=512, gathered A)
    {
        int tiles = (NNODES / 16) * (PDIM / 16);     // 2048
        wmma_gemm_abt<<<tiles / 8, 256, 0, stream>>>(
            emb, INFEAT, proj_w, PDIM, Hn, PDIM,
            NNODES / 16, PDIM / 16, INFEAT, x);
    }
    // K2: gsh = Hn @ lin_w^T                (M=512, N=256, K=1024)
    {
        int tiles = (NNODES / 16) * (RDIM / 16);     // 512
        wmma_gemm_abt<<<tiles / 8, 256, 0, stream>>>(
            Hn, PDIM, lin_w, PDIM, gsh, RDIM,
            NNODES / 16, RDIM / 16, PDIM, nullptr);
    }
    // K3: conditioning path (tiny matvecs)
    cond_proj<<<(BATCH * PDIM) / 256, 256, 0, stream>>>(proj_w, proj_b, cond, t1);
    cond_lin <<<(BATCH * RDIM) / 256, 256, 0, stream>>>(lin_w, t1, cvec);
    // K4: broadcast add -> g[b][i][r]
    add_cond<<<(BATCH * NNODES * RDIM) / 256, 256, 0, stream>>>(gsh, cvec, gg);
    // K5: fused attention (scores + masked softmax + WMMA aggregation)
    size_t shbytes = (size_t)(NNODES * GH_STRIDE + 16 * S_STRIDE + FDIM) * sizeof(float);
    gat_attn<<<dim3(NNODES / 16, NHEADS, BATCH), 512, shbytes, stream>>>(
        gg, adj, attn_w, out);
}